// RelGraphConvLayer_54305566491324
// MI455X (gfx1250) — compile-verified
//
#include <hip/hip_runtime.h>
#include <hip/hip_bf16.h>
#include <stdint.h>

typedef __attribute__((ext_vector_type(2))) float v2f;
typedef __attribute__((ext_vector_type(8))) float v8f;

#define TM  32    // output rows per block
#define KC  64    // K chunk staged through LDS
#define LDP 66    // padded LDS row stride (floats) -> conflict-free b64 reads

// -------------------------------------------------------------------------
// GEMM: H[M x 256] = X[M x 256] * W[256 x 256], fp32 via V_WMMA_F32_16X16X4_F32
// Block = 256 threads = 8 waves. Wave w: rows m_off = 16*(w/4), cols n_off = 64*(w%4).
// -------------------------------------------------------------------------
__global__ __launch_bounds__(256) void rgcn_gemm(const float* __restrict__ X,
                                                 const float* __restrict__ W,
                                                 float* __restrict__ H,
                                                 int M) {
    __shared__ float sA[TM][LDP];     // 32 x 66 x 4B  = 8448 B
    __shared__ float sB[256][LDP];    // 256 x 66 x 4B = 67584 B  (W^T chunk: [n][k])

    const int tid   = threadIdx.x;
    const int lane  = tid & 31;
    const int wave  = tid >> 5;
    const int m_off = (wave >> 2) << 4;   // 0 or 16
    const int n_off = (wave & 3) << 6;    // 0,64,128,192
    const int m0    = blockIdx.x * TM;
    if (m0 + TM > M) return;              // M divisible by 32 for this problem

    const int half = lane >> 4;           // lane half selects K pair (ISA 7.12.2)
    const int mr   = lane & 15;

    v8f acc0 = {}, acc1 = {}, acc2 = {}, acc3 = {};

    for (int kc = 0; kc < 256; kc += KC) {
        // ---- stage A chunk (32 x 64) with async global->LDS (ASYNCcnt path) ----
        #pragma unroll
        for (int i = 0; i < 2; ++i) {
            int l   = tid + i * 256;              // [0, 512) float4 units
            int row = l >> 4;                     // 16 float4 per row
            int kk  = (l & 15) << 2;
            unsigned lds_off = (unsigned)(uintptr_t)(&sA[row][kk]);
            unsigned long long gaddr =
                (unsigned long long)(uintptr_t)(X + (size_t)(m0 + row) * 256 + kc + kk);
            asm volatile("global_load_async_to_lds_b128 %0, %1, off"
                         :: "v"(lds_off), "v"(gaddr)
                         : "memory");
        }

        // ---- stage B chunk transposed: sB[n][k] = W[kc+k][n] ----
        #pragma unroll
        for (int i = 0; i < 16; ++i) {
            int l  = (tid + i * 256) << 2;        // float index in [0, 16384)
            int kr = l >> 8;                      // 0..63
            int nc = l & 255;
            float4 v = *(const float4*)(W + (size_t)(kc + kr) * 256 + nc);
            sB[nc + 0][kr] = v.x;
            sB[nc + 1][kr] = v.y;
            sB[nc + 2][kr] = v.z;
            sB[nc + 3][kr] = v.w;
        }

        asm volatile("s_wait_asynccnt 0" ::: "memory");
        __syncthreads();

        // ---- WMMA inner loop over the K chunk ----
        #pragma unroll
        for (int ks = 0; ks < KC; ks += 4) {
            const int ka = ks + (half << 1);      // this lane-half's K pair
            v2f a  = *(const v2f*)&sA[m_off + mr][ka];
            v2f b0 = *(const v2f*)&sB[n_off +  0 + mr][ka];
            v2f b1 = *(const v2f*)&sB[n_off + 16 + mr][ka];
            v2f b2 = *(const v2f*)&sB[n_off + 32 + mr][ka];
            v2f b3 = *(const v2f*)&sB[n_off + 48 + mr][ka];
            acc0 = __builtin_amdgcn_wmma_f32_16x16x4_f32(false, a, false, b0,
                                                         (short)0, acc0, false, false);
            acc1 = __builtin_amdgcn_wmma_f32_16x16x4_f32(false, a, false, b1,
                                                         (short)0, acc1, false, false);
            acc2 = __builtin_amdgcn_wmma_f32_16x16x4_f32(false, a, false, b2,
                                                         (short)0, acc2, false, false);
            acc3 = __builtin_amdgcn_wmma_f32_16x16x4_f32(false, a, false, b3,
                                                         (short)0, acc3, false, false);
        }
        __syncthreads();
    }

    // ---- store accumulators; C layout: VGPR v -> M = m_off + v + 8*half, N = lane%16 ----
    #pragma unroll
    for (int v = 0; v < 8; ++v) {
        size_t row = (size_t)(m0 + m_off + (half << 3) + v) * 256;
        H[row + n_off + mr +  0] = acc0[v];
        H[row + n_off + mr + 16] = acc1[v];
        H[row + n_off + mr + 32] = acc2[v];
        H[row + n_off + mr + 48] = acc3[v];
    }
}

// -------------------------------------------------------------------------
// Scatter: out[dst[e]] += h[src[e]]  (256 floats/edge, 64 threads/edge, float4)
// -------------------------------------------------------------------------
__global__ __launch_bounds__(256) void rgcn_scatter(const float* __restrict__ h,
                                                    const long long* __restrict__ src,
                                                    const long long* __restrict__ dst,
                                                    float* __restrict__ out,
                                                    int n_edges) {
    const int t = threadIdx.x;
    const long long e = (long long)blockIdx.x * 4 + (t >> 6);
    if (e >= n_edges) return;
    const int c = (t & 63) << 2;
    const long long s = src[e];
    const long long d = dst[e];
    const float4 v = *(const float4*)(h + (size_t)s * 256 + c);
    float* o = out + (size_t)d * 256 + c;
    atomicAdd(o + 0, v.x);
    atomicAdd(o + 1, v.y);
    atomicAdd(o + 2, v.z);
    atomicAdd(o + 3, v.w);
}

// -------------------------------------------------------------------------
// In-place ReLU (float4)
// -------------------------------------------------------------------------
__global__ __launch_bounds__(256) void rgcn_relu(float* __restrict__ out, long long n4) {
    long long i = (long long)blockIdx.x * blockDim.x + threadIdx.x;
    if (i >= n4) return;
    float4* p = (float4*)out + i;
    float4 v = *p;
    v.x = fmaxf(v.x, 0.f);
    v.y = fmaxf(v.y, 0.f);
    v.z = fmaxf(v.z, 0.f);
    v.w = fmaxf(v.w, 0.f);
    *p = v;
}

extern "C" void kernel_launch(void* const* d_in, const int* in_sizes, int n_in,
                              void* d_out, int out_size, void* d_ws, size_t ws_size,
                              hipStream_t stream) {
    const float*     x   = (const float*)d_in[0];
    const float*     w   = (const float*)d_in[1];
    const long long* src = (const long long*)d_in[2];
    const long long* dst = (const long long*)d_in[3];
    float* out = (float*)d_out;
    float* h   = (float*)d_ws;   // one relation's h: n_nodes*256 floats (~100 MB, L2-resident)

    const int K = 256, N = 256;
    const int n_nodes = in_sizes[0] / K;            // 100000
    const int n_rel   = in_sizes[1] / (K * N);      // 3
    const int n_edges = in_sizes[2] / n_rel;        // 300000

    // zero the fp32 accumulator (graph-capturable memset node)
    hipMemsetAsync(d_out, 0, (size_t)out_size * sizeof(float), stream);

    dim3 blk(256);
    const int gemm_grid = (n_nodes + TM - 1) / TM;  // 3125
    const int scat_grid = (n_edges + 3) / 4;        // 75000

    // Process relations serially: h_r stays hot in the 192MB L2 between
    // the GEMM that produces it and the scatter that randomly gathers it.
    for (int r = 0; r < n_rel; ++r) {
        rgcn_gemm<<<gemm_grid, blk, 0, stream>>>(x, w + (size_t)r * K * N, h, n_nodes);
        rgcn_scatter<<<scat_grid, blk, 0, stream>>>(h, src + (size_t)r * n_edges,
                                                    dst + (size_t)r * n_edges,
                                                    out, n_edges);
    }

    const long long n4 = (long long)out_size / 4;
    const int relu_grid = (int)((n4 + 255) / 256);
    rgcn_relu<<<relu_grid, blk, 0, stream>>>(out, n4);
}